// LinOSSBlock_22368189677869
// MI455X (gfx1250) — compile-verified
//
#include <hip/hip_runtime.h>
#include <hip/hip_bf16.h>

// ---------------------------------------------------------------------------
// LinOSS block for MI455X (gfx1250, wave32).
//   P = H = 256, BATCH = 8, L = 4096, M = BATCH*L = 32768
// Pipeline:
//   1) prep_params : dt_s, m11, m12 (soft projection of A/G/dt)
//   2) pack_BC     : pair-interleaved K-major packs of B (re/im), C (re/-im)
//                    Bp[(k&~1)*P + 2n + (k&1)] -> one v2f = lane fragment
//   3) gemm_bu     : f = dt_s * (u @ B^T)   via v_wmma_f32_16x16x4_f32
//   4) scan_chunks : per-chunk local scan (zero init) -> carries
//   5) scan_combine: M^256 by squaring, serial combine of 16 carries
//   6) scan_apply  : re-scan chunks with true init; ys overwrites f in place
//   7) gemm_out    : out = ys_re@Cre + ys_im@(-Cim) + D*u  via WMMA
// ---------------------------------------------------------------------------

typedef float v2f __attribute__((ext_vector_type(2)));
typedef float v8f __attribute__((ext_vector_type(8)));

#define PDIM 256
#define HDIM 256
#define NBATCH 8
#define LSEQ 4096
#define MROWS (NBATCH * LSEQ)   // 32768
#define LC 256                  // scan chunk length
#define NCHUNK (LSEQ / LC)      // 16
// pitch % 64 == 4: lane bank = 4*mn + 2*hi -> 32 distinct banks, and
// (mn*260 + ksel + k) is even -> 8-byte aligned ds_load_b64 fragments.
#define LDSPITCH 260

// f32 WMMA: D(16x16,f32) = A(16x4,f32) x B(4x16,f32) + C
static __device__ inline v8f wmma4f32(v2f a, v2f b, v8f c) {
  return __builtin_amdgcn_wmma_f32_16x16x4_f32(
      /*neg_a=*/false, a, /*neg_b=*/false, b,
      /*c_mod=*/(short)0, c, /*reuse_a=*/false, /*reuse_b=*/false);
}

// ------------------------- 1) scalar parameter prep -------------------------
__global__ void prep_params(const float* __restrict__ A_diag,
                            const float* __restrict__ G_diag,
                            const float* __restrict__ dt,
                            float* __restrict__ m11,
                            float* __restrict__ m12,
                            float* __restrict__ dts) {
  int p = threadIdx.x;                      // 256 threads
  float s = 1.0f / (1.0f + __expf(-dt[p])); // sigmoid
  float Glow = s * A_diag[p];
  float G = Glow + fmaxf(G_diag[p] - Glow, 0.0f);
  float Alow = 0.25f * G * G;
  float A = Alow + fmaxf(A_diag[p] - Alow, 0.0f);
  m11[p] = 1.0f - s * G;   // m21 = s, m22 = 1
  m12[p] = -s * A;
  dts[p] = s;
}

// ---------------- 2) pack B / C, pair-interleaved K-major -------------------
// B operand (K=h, N=p):  BreP[(h&~1)*P + 2p + (h&1)] =  B[p,h,0]
// C operand (K=p, N=h):  CreP[(p&~1)*H + 2h + (p&1)] =  C[h,p,0]
//                        CimN[(p&~1)*H + 2h + (p&1)] = -C[h,p,1]
__global__ void pack_BC(const float* __restrict__ Bmat,
                        const float* __restrict__ Cmat,
                        float* __restrict__ BreP, float* __restrict__ BimP,
                        float* __restrict__ CreP, float* __restrict__ CimN) {
  int idx = blockIdx.x * blockDim.x + threadIdx.x;  // 65536
  int p = idx & (PDIM - 1);
  int h = idx >> 8;
  float br = Bmat[(p * HDIM + h) * 2 + 0];
  float bi = Bmat[(p * HDIM + h) * 2 + 1];
  int bpos = (h & ~1) * PDIM + 2 * p + (h & 1);
  BreP[bpos] = br;
  BimP[bpos] = bi;
  float cr = Cmat[(h * PDIM + p) * 2 + 0];
  float ci = Cmat[(h * PDIM + p) * 2 + 1];
  int cpos = (p & ~1) * HDIM + 2 * h + (p & 1);
  CreP[cpos] = cr;
  CimN[cpos] = -ci;
}

// --------------------------- 3) input GEMM (WMMA) ---------------------------
// block = 8 waves; block tile = 16 rows x 128 p; each wave: 16x16, re+im.
__global__ __launch_bounds__(256) void gemm_bu(
    const float* __restrict__ u, const float* __restrict__ BreP,
    const float* __restrict__ BimP, const float* __restrict__ dts,
    float* __restrict__ fre, float* __restrict__ fim) {
  __shared__ float su[16 * LDSPITCH];
  const int tid = threadIdx.x;
  const int rowBase = blockIdx.x * 16;

  // A-tile: 16 consecutive rows of u = 4096 contiguous floats
  const float* src = u + (size_t)rowBase * HDIM;
  for (int i = tid; i < 16 * HDIM; i += 256)
    su[(i >> 8) * LDSPITCH + (i & 255)] = src[i];
  __syncthreads();

  const int wave = tid >> 5, lane = tid & 31;
  const int pBase = blockIdx.y * 128 + wave * 16;
  const int mn = lane & 15;              // A: row m, B: col n
  const int ksel = (lane >> 4) << 1;     // lanes 0-15: K 0,1; 16-31: K 2,3

  v8f accRe = {};
  v8f accIm = {};
  const float* arow = su + mn * LDSPITCH + ksel;   // even offset -> b64 loads
  const float* brow = BreP + 2 * (pBase + mn);
  const float* irow = BimP + 2 * (pBase + mn);

  for (int k = 0; k < HDIM; k += 4) {
    v2f a = *(const v2f*)(arow + k);                       // ds_load_b64
    v2f b0 = *(const v2f*)(brow + (k + ksel) * PDIM);      // global_load_b64
    v2f b1 = *(const v2f*)(irow + (k + ksel) * PDIM);
    accRe = wmma4f32(a, b0, accRe);
    accIm = wmma4f32(a, b1, accIm);
  }

  const float scale = dts[pBase + mn];   // f1 = dt_s * Bu
  const int mo = rowBase + ((lane >> 4) << 3);
  const int col = pBase + mn;
#pragma unroll
  for (int r = 0; r < 8; ++r) {
    fre[(size_t)(mo + r) * PDIM + col] = accRe[r] * scale;
    fim[(size_t)(mo + r) * PDIM + col] = accIm[r] * scale;
  }
}

// ----------------------- 4) chunk-local scan (zero init) --------------------
__global__ void scan_chunks(const float* __restrict__ fre,
                            const float* __restrict__ fim,
                            float* __restrict__ c1, float* __restrict__ c2,
                            const float* __restrict__ m11,
                            const float* __restrict__ m12,
                            const float* __restrict__ dts) {
  int tid = blockIdx.x * blockDim.x + threadIdx.x;  // 65536
  int p = tid & (PDIM - 1);
  int comp = (tid >> 8) & 1;
  int chunk = (tid >> 9) & (NCHUNK - 1);
  int b = tid >> 13;

  const float* f = (comp ? fim : fre) +
                   (size_t)(b * LSEQ + chunk * LC) * PDIM + p;
  const float a11 = m11[p], a12 = m12[p], a21 = dts[p];
  float s1 = 0.0f, s2 = 0.0f;
  for (int t = 0; t < LC; ++t) {
    float ft = f[(size_t)t * PDIM];
    float n1 = fmaf(a11, s1, fmaf(a12, s2, ft));
    float n2 = fmaf(a21, s1, s2);
    s1 = n1;
    s2 = n2;
  }
  int ci = ((b * 2 + comp) * NCHUNK + chunk) * PDIM + p;
  c1[ci] = s1;
  c2[ci] = s2;
}

// ------------- 5) combine carries: M^LC by squaring + serial pass -----------
__global__ void scan_combine(float* __restrict__ c1, float* __restrict__ c2,
                             const float* __restrict__ m11,
                             const float* __restrict__ m12,
                             const float* __restrict__ dts) {
  int tid = blockIdx.x * blockDim.x + threadIdx.x;  // 4096
  int p = tid & (PDIM - 1);
  int comp = (tid >> 8) & 1;
  int b = tid >> 9;

  // M^256: 8 squarings of the per-p 2x2
  float a = m11[p], bb = m12[p], c = dts[p], d = 1.0f;
#pragma unroll
  for (int i = 0; i < 8; ++i) {
    float na = fmaf(a, a, bb * c);
    float nb = fmaf(a, bb, bb * d);
    float nc = fmaf(c, a, d * c);
    float nd = fmaf(c, bb, d * d);
    a = na; bb = nb; c = nc; d = nd;
  }

  float i1 = 0.0f, i2 = 0.0f;  // init state for chunk 0
  int base = (b * 2 + comp) * NCHUNK * PDIM + p;
  for (int ch = 0; ch < NCHUNK; ++ch) {
    int ci = base + ch * PDIM;
    float s1 = c1[ci], s2 = c2[ci];     // chunk-local affine part
    c1[ci] = i1;                         // overwrite with init state
    c2[ci] = i2;
    float t1 = fmaf(a, i1, fmaf(bb, i2, s1));
    float t2 = fmaf(c, i1, fmaf(d, i2, s2));
    i1 = t1;
    i2 = t2;
  }
}

// ------------- 6) re-scan with true init; ys=x2 overwrites f ---------------
__global__ void scan_apply(float* __restrict__ fre, float* __restrict__ fim,
                           const float* __restrict__ c1,
                           const float* __restrict__ c2,
                           const float* __restrict__ m11,
                           const float* __restrict__ m12,
                           const float* __restrict__ dts) {
  int tid = blockIdx.x * blockDim.x + threadIdx.x;  // 65536
  int p = tid & (PDIM - 1);
  int comp = (tid >> 8) & 1;
  int chunk = (tid >> 9) & (NCHUNK - 1);
  int b = tid >> 13;

  float* f = (comp ? fim : fre) + (size_t)(b * LSEQ + chunk * LC) * PDIM + p;
  const float a11 = m11[p], a12 = m12[p], a21 = dts[p];
  int ci = ((b * 2 + comp) * NCHUNK + chunk) * PDIM + p;
  float x1 = c1[ci], x2 = c2[ci];
  for (int t = 0; t < LC; ++t) {
    float ft = f[(size_t)t * PDIM];
    float y2 = fmaf(a21, x1, x2);                   // x2_t (m22 = 1, f2 = 0)
    float y1 = fmaf(a11, x1, fmaf(a12, x2, ft));    // x1_t
    f[(size_t)t * PDIM] = y2;                       // ys = x2, in place
    x1 = y1;
    x2 = y2;
  }
}

// --------------------------- 7) output GEMM (WMMA) --------------------------
// out = ys_re @ Cre + ys_im @ (-Cim) + D*u
__global__ __launch_bounds__(256) void gemm_out(
    const float* __restrict__ ysre, const float* __restrict__ ysim,
    const float* __restrict__ CreP, const float* __restrict__ CimN,
    const float* __restrict__ u, const float* __restrict__ Dvec,
    float* __restrict__ out) {
  __shared__ float sre[16 * LDSPITCH];
  __shared__ float sim[16 * LDSPITCH];
  const int tid = threadIdx.x;
  const int rowBase = blockIdx.x * 16;

  const float* pr = ysre + (size_t)rowBase * PDIM;
  const float* pi = ysim + (size_t)rowBase * PDIM;
  for (int i = tid; i < 16 * PDIM; i += 256) {
    int r = i >> 8, c = i & 255;
    sre[r * LDSPITCH + c] = pr[i];
    sim[r * LDSPITCH + c] = pi[i];
  }
  __syncthreads();

  const int wave = tid >> 5, lane = tid & 31;
  const int hBase = blockIdx.y * 128 + wave * 16;
  const int mn = lane & 15;
  const int ksel = (lane >> 4) << 1;

  v8f acc = {};
  const float* arow_re = sre + mn * LDSPITCH + ksel;
  const float* arow_im = sim + mn * LDSPITCH + ksel;
  const float* crow = CreP + 2 * (hBase + mn);
  const float* irow = CimN + 2 * (hBase + mn);

  for (int k = 0; k < PDIM; k += 4) {
    v2f ar = *(const v2f*)(arow_re + k);                   // ds_load_b64
    v2f ai = *(const v2f*)(arow_im + k);
    v2f vr = *(const v2f*)(crow + (k + ksel) * HDIM);      // global_load_b64
    v2f vi = *(const v2f*)(irow + (k + ksel) * HDIM);
    acc = wmma4f32(ar, vr, acc);   // + ys_re * Cre
    acc = wmma4f32(ai, vi, acc);   // + ys_im * (-Cim)
  }

  const int h = hBase + mn;
  const float dh = Dvec[h];
  const int mo = rowBase + ((lane >> 4) << 3);
#pragma unroll
  for (int r = 0; r < 8; ++r) {
    size_t idx = (size_t)(mo + r) * HDIM + h;
    out[idx] = acc[r] + dh * u[idx];
  }
}

// ---------------------------------------------------------------------------
extern "C" void kernel_launch(void* const* d_in, const int* in_sizes, int n_in,
                              void* d_out, int out_size, void* d_ws,
                              size_t ws_size, hipStream_t stream) {
  const float* u      = (const float*)d_in[0];   // (8, 4096, 256)
  const float* A_diag = (const float*)d_in[1];   // (256,)
  const float* G_diag = (const float*)d_in[2];   // (256,)
  const float* dt     = (const float*)d_in[3];   // (256,)
  const float* Bmat   = (const float*)d_in[4];   // (256, 256, 2)
  const float* Cmat   = (const float*)d_in[5];   // (256, 256, 2)
  const float* Dvec   = (const float*)d_in[6];   // (256,)
  float* out = (float*)d_out;

  float* w = (float*)d_ws;
  const size_t NF = (size_t)MROWS * PDIM;        // 8,388,608
  float* BreP = w;
  float* BimP = BreP + (size_t)HDIM * PDIM;
  float* CreP = BimP + (size_t)HDIM * PDIM;
  float* CimN = CreP + (size_t)PDIM * HDIM;
  float* m11  = CimN + (size_t)PDIM * HDIM;
  float* m12  = m11 + PDIM;
  float* dts  = m12 + PDIM;
  float* fre  = dts + PDIM;                      // 8-byte aligned (even offset)
  float* fim  = fre + NF;
  float* c1   = fim + NF;                        // carries: 65536 each
  float* c2   = c1 + (size_t)NBATCH * 2 * NCHUNK * PDIM;

  prep_params<<<1, 256, 0, stream>>>(A_diag, G_diag, dt, m11, m12, dts);
  pack_BC<<<256, 256, 0, stream>>>(Bmat, Cmat, BreP, BimP, CreP, CimN);
  gemm_bu<<<dim3(MROWS / 16, PDIM / 128), 256, 0, stream>>>(
      u, BreP, BimP, dts, fre, fim);
  scan_chunks<<<256, 256, 0, stream>>>(fre, fim, c1, c2, m11, m12, dts);
  scan_combine<<<16, 256, 0, stream>>>(c1, c2, m11, m12, dts);
  scan_apply<<<256, 256, 0, stream>>>(fre, fim, c1, c2, m11, m12, dts);
  gemm_out<<<dim3(MROWS / 16, HDIM / 128), 256, 0, stream>>>(
      fre, fim, CreP, CimN, u, Dvec, out);
}